// SimpleAnomalyGenerator_72404558676538
// MI455X (gfx1250) — compile-verified
//
#include <hip/hip_runtime.h>

// Problem constants (match reference)
#define BB 128
#define NB 5
#define CC 3
#define HH 256
#define WW 256
#define BETA_LO 0.1f
#define BETA_HI 1.0f

typedef float v4f __attribute__((ext_vector_type(4)));

// Block = 256 threads (8 wave32) covers 4 rows of one batch image.
// Each thread handles 4 consecutive pixels -> B128 vector memory ops.
#define ROWS_PER_BLOCK 4

__global__ __launch_bounds__(256)
void anomaly_gen_kernel(const float* __restrict__ images,
                        const float* __restrict__ blend,
                        const float* __restrict__ beta_u,
                        const int*   __restrict__ cxs,
                        const int*   __restrict__ cys,
                        const int*   __restrict__ rads,
                        float* __restrict__ out_aug,
                        float* __restrict__ out_mask)
{
    const int blk  = blockIdx.x;
    const int b    = blk / (HH / ROWS_PER_BLOCK);        // uniform per block
    const int yblk = blk % (HH / ROWS_PER_BLOCK);
    const int y    = yblk * ROWS_PER_BLOCK + (threadIdx.x >> 6);
    const int x0   = (threadIdx.x & 63) << 2;            // 0..252 step 4

    // Blob parameters: uniform per block -> compiler emits s_load (SMEM path).
    int cx[NB], cy[NB], r2[NB];
#pragma unroll
    for (int i = 0; i < NB; ++i) {
        cx[i] = cxs[b * NB + i];
        cy[i] = cys[b * NB + i];
        const int r = rads[b * NB + i];
        r2[i] = r * r;
    }
    const float beta = beta_u[b] * (BETA_HI - BETA_LO) + BETA_LO;

    // Integer circle test == reference's sqrt(float) <= r (exact, see analysis).
    bool an[4];
#pragma unroll
    for (int p = 0; p < 4; ++p) {
        const int x = x0 + p;
        bool a = false;
#pragma unroll
        for (int i = 0; i < NB; ++i) {
            const int dx = x - cx[i];
            const int dy = y - cy[i];
            a |= (dx * dx + dy * dy) <= r2[i];
        }
        an[p] = a;
    }
    const bool anyAn = an[0] | an[1] | an[2] | an[3];
    const bool anyCl = !(an[0] & an[1] & an[2] & an[3]);

    const size_t pix = (size_t)y * WW + (size_t)x0;

    // Anomaly mask output (streamed once -> non-temporal store).
    v4f m;
    m.x = an[0] ? 1.0f : 0.0f;
    m.y = an[1] ? 1.0f : 0.0f;
    m.z = an[2] ? 1.0f : 0.0f;
    m.w = an[3] ? 1.0f : 0.0f;
    __builtin_nontemporal_store(m, (v4f*)(out_mask + (size_t)b * HH * WW + pix));

    // gfx1250 prefetch path: hint next 4-row tile of channel 0 into L2.
    __builtin_prefetch(images + ((size_t)b * CC) * HH * WW + pix + ROWS_PER_BLOCK * WW, 0, 1);

#pragma unroll
    for (int c = 0; c < CC; ++c) {
        const size_t off = ((size_t)b * CC + c) * (size_t)(HH * WW) + pix;
        v4f img = {0.f, 0.f, 0.f, 0.f};
        v4f bl  = {0.f, 0.f, 0.f, 0.f};
        if (anyCl) img = __builtin_nontemporal_load((const v4f*)(images + off));
        if (anyAn) bl  = __builtin_nontemporal_load((const v4f*)(blend  + off));

        v4f o;
        o.x = an[0] ? fminf(fmaxf(bl.x * beta, 0.0f), 1.0f) : fminf(fmaxf(img.x, 0.0f), 1.0f);
        o.y = an[1] ? fminf(fmaxf(bl.y * beta, 0.0f), 1.0f) : fminf(fmaxf(img.y, 0.0f), 1.0f);
        o.z = an[2] ? fminf(fmaxf(bl.z * beta, 0.0f), 1.0f) : fminf(fmaxf(img.z, 0.0f), 1.0f);
        o.w = an[3] ? fminf(fmaxf(bl.w * beta, 0.0f), 1.0f) : fminf(fmaxf(img.w, 0.0f), 1.0f);
        __builtin_nontemporal_store(o, (v4f*)(out_aug + off));
    }
}

extern "C" void kernel_launch(void* const* d_in, const int* in_sizes, int n_in,
                              void* d_out, int out_size, void* d_ws, size_t ws_size,
                              hipStream_t stream)
{
    // setup_inputs() order:
    // 0: images      [B,C,H,W] f32
    // 1: blob_noise  [B,NB,H,W] f32  -- provably unused (see analysis)
    // 2: blend_noise [B,C,H,W] f32
    // 3: beta_u      [B,1,1,1] f32
    // 4: centers_x   [B,NB] i32
    // 5: centers_y   [B,NB] i32
    // 6: radii       [B,NB] i32
    const float* images = (const float*)d_in[0];
    const float* blend  = (const float*)d_in[2];
    const float* beta_u = (const float*)d_in[3];
    const int*   cxs    = (const int*)d_in[4];
    const int*   cys    = (const int*)d_in[5];
    const int*   rads   = (const int*)d_in[6];

    float* out_aug  = (float*)d_out;                                  // B*C*H*W
    float* out_mask = (float*)d_out + (size_t)BB * CC * HH * WW;      // B*1*H*W

    const int grid = BB * (HH / ROWS_PER_BLOCK);   // 128 * 64 = 8192 blocks
    anomaly_gen_kernel<<<grid, 256, 0, stream>>>(images, blend, beta_u,
                                                 cxs, cys, rads,
                                                 out_aug, out_mask);
}